// BiBoMoELayer_15333033247084
// MI455X (gfx1250) — compile-verified
//
#include <hip/hip_runtime.h>
#include <math.h>

// ---- problem constants ----
#define T_TOK   16384      // B*S
#define H_DIM   1024
#define I_DIM   512
#define E_NUM   16
#define E_R     15         // routed (non-identity) experts
#define S_LEN   4096
#define KS      3
#define HI      (H_DIM * I_DIM)

typedef float v2f __attribute__((ext_vector_type(2)));
typedef float v8f __attribute__((ext_vector_type(8)));

static __device__ __forceinline__ v8f wmma_f32(v2f a, v2f b, v8f c) {
  // V_WMMA_F32_16X16X4_F32 : D = A(16x4) * B(4x16) + C(16x16)
  return __builtin_amdgcn_wmma_f32_16x16x4_f32(false, a, false, b, (short)0, c,
                                               false, false);
}

static __device__ __forceinline__ v8f vzero8() {
  v8f v = {0.f, 0.f, 0.f, 0.f, 0.f, 0.f, 0.f, 0.f};
  return v;
}

// Low 32 bits of a generic pointer derived from LDS = wave-relative LDS offset.
static __device__ __forceinline__ unsigned lds_off(const void* p) {
  return (unsigned)(uintptr_t)p;
}

// GLOBAL_LOAD_ASYNC_TO_LDS_B128: copies 16 bytes from the global address in vaddr to the LDS address in vdst, tracked by ASYNCcnt.
static __device__ __forceinline__ void async_load_b128(unsigned dst_lds,
                                                       const float* gptr) {
  asm volatile("global_load_async_to_lds_b128 %0, %1, off"
               :: "v"(dst_lds), "v"((unsigned long long)(uintptr_t)gptr)
               : "memory");
}

static __device__ __forceinline__ void wait_async0() {
#if __has_builtin(__builtin_amdgcn_s_wait_asynccnt)
  __builtin_amdgcn_s_wait_asynccnt(0);
#else
  asm volatile("s_wait_asynccnt 0x0" ::: "memory");
#endif
}

// A-fragment (16x4) from LDS row-major tile with stride LD (floats).
// lane l holds A[row][kb + 2*(l>>4) + {0,1}].
static __device__ __forceinline__ v2f load_afrag(const float* lds, int LD,
                                                 int row, int kb, int lane) {
  int half = lane >> 4;
  const float2* p = (const float2*)(lds + row * LD + kb + 2 * half);
  float2 t = *p;
  v2f a; a.x = t.x; a.y = t.y;
  return a;
}

// B-fragment (4x16) from global row-major W (K x N), ld = N.
// lane l holds W[k0 + 2*(l>>4) + {0,1}][n0 + (l&15)].
static __device__ __forceinline__ v2f load_bfrag(const float* __restrict__ W,
                                                 int ld, int k0, int n0, int lane) {
  int half = lane >> 4;
  int col  = lane & 15;
  const float* p = W + (size_t)(k0 + 2 * half) * ld + n0 + col;
  v2f b; b.x = p[0]; b.y = p[ld];
  return b;
}

static __device__ __forceinline__ float silu(float x) {
  return x / (1.f + expf(-x));
}

// ------------------------------------------------------------------
// 0) zero the per-expert counters
// ------------------------------------------------------------------
__global__ void init_counts_kernel(int* counts) {
  if (threadIdx.x < E_NUM) counts[threadIdx.x] = 0;
}

// ------------------------------------------------------------------
// 1) conv_w (I,H,KS) -> 3 row-major (H x I) matrices for GEMM B-loads
// ------------------------------------------------------------------
__global__ void conv_transpose_kernel(const float* __restrict__ cw,
                                      float* __restrict__ wct) {
  int idx = blockIdx.x * blockDim.x + threadIdx.x;     // over H*I
  if (idx >= HI) return;
  int h = idx / I_DIM;
  int i = idx - h * I_DIM;
  const float* p = cw + ((size_t)i * H_DIM + h) * KS;
  wct[0 * HI + h * I_DIM + i] = p[0];
  wct[1 * HI + h * I_DIM + i] = p[1];
  wct[2 * HI + h * I_DIM + i] = p[2];
}

// ------------------------------------------------------------------
// 2) router: wave-per-token logits -> softmax -> top2 -> buckets
// ------------------------------------------------------------------
__global__ __launch_bounds__(256)
void router_kernel(const float* __restrict__ x, const float* __restrict__ rw,
                   const float* __restrict__ rb, int* counts, float* id_w,
                   int* etok, float* ewt) {
  int wave = threadIdx.x >> 5;
  int lane = threadIdx.x & 31;
  int t = blockIdx.x * 8 + wave;

  float acc[E_NUM];
#pragma unroll
  for (int e = 0; e < E_NUM; ++e) acc[e] = 0.f;

  const float* xr = x + (size_t)t * H_DIM;
  for (int j = 0; j < H_DIM / 32; ++j) {
    int h = j * 32 + lane;
    float xv = xr[h];
    const float* wr = rw + (size_t)h * E_NUM;
#pragma unroll
    for (int e = 0; e < E_NUM; ++e) acc[e] += xv * wr[e];
  }
#pragma unroll
  for (int e = 0; e < E_NUM; ++e) {
#pragma unroll
    for (int off = 16; off > 0; off >>= 1) acc[e] += __shfl_xor(acc[e], off, 32);
  }

  if (lane == 0) {
    float lg[E_NUM];
#pragma unroll
    for (int e = 0; e < E_NUM; ++e) lg[e] = acc[e] + rb[e];
    float mx = lg[0];
#pragma unroll
    for (int e = 1; e < E_NUM; ++e) mx = fmaxf(mx, lg[e]);
    float p[E_NUM], se = 0.f;
#pragma unroll
    for (int e = 0; e < E_NUM; ++e) { p[e] = expf(lg[e] - mx); se += p[e]; }
    float inv_se = 1.f / se;
#pragma unroll
    for (int e = 0; e < E_NUM; ++e) p[e] *= inv_se;

    int i0 = 0; float p0 = p[0];
#pragma unroll
    for (int e = 1; e < E_NUM; ++e) if (p[e] > p0) { p0 = p[e]; i0 = e; }
    int i1 = (i0 == 0) ? 1 : 0; float p1 = p[i1];
#pragma unroll
    for (int e = 0; e < E_NUM; ++e)
      if (e != i0 && p[e] > p1) { p1 = p[e]; i1 = e; }

    float invn = 1.f / (p0 + p1 + 1e-6f);
    float w0 = p0 * invn, w1 = p1 * invn;

    float idw = 0.f;
    if (i0 == E_NUM - 1) idw = w0;
    else { int s = atomicAdd(&counts[i0], 1); etok[i0 * T_TOK + s] = t; ewt[i0 * T_TOK + s] = w0; }
    if (i1 == E_NUM - 1) idw = w1;
    else { int s = atomicAdd(&counts[i1], 1); etok[i1 * T_TOK + s] = t; ewt[i1 * T_TOK + s] = w1; }
    id_w[t] = idw;
  }
}

// ------------------------------------------------------------------
// 3) shared expert, M=32 tile: conv-gate (3 shifted GEMMs) * up, then down.
//    X chunks double-buffered via GLOBAL_LOAD_ASYNC_TO_LDS_B128.
//    Writes out = shared + id_w * x  (base for expert atomics).
// ------------------------------------------------------------------
__global__ __launch_bounds__(256)
void shared_kernel(const float* __restrict__ x, const float* __restrict__ wct,
                   const float* __restrict__ sup, const float* __restrict__ sdn,
                   const float* __restrict__ id_w, float* __restrict__ out) {
  __shared__ float xs[2][34 * 68];  // 32+2 halo rows, stride 68 (==4 mod 64)
  __shared__ float act[32 * 516];   // silu(gate)*up, stride 516 (==4 mod 64)

  int t0 = blockIdx.x * 32;
  int s0 = t0 & (S_LEN - 1);
  int lane = threadIdx.x & 31;
  int wv   = threadIdx.x >> 5;
  int mrow = lane & 15;

  v8f gacc[4][2], uacc[4][2];
#pragma unroll
  for (int nt = 0; nt < 4; ++nt)
#pragma unroll
    for (int mt = 0; mt < 2; ++mt) { gacc[nt][mt] = vzero8(); uacc[nt][mt] = vzero8(); }
  int nb = wv * 64;   // this wave's 64 columns of I

  // prologue: stage chunk 0
  {
    float* buf = xs[0];
    for (int idx = threadIdx.x; idx < 34 * 16; idx += 256) {
      int r = idx >> 4, c4 = idx & 15;
      if (s0 + r >= 2)
        async_load_b128(lds_off(buf + r * 68 + c4 * 4),
                        x + (size_t)(t0 - 2 + r) * H_DIM + c4 * 4);
      else
        *(float4*)(buf + r * 68 + c4 * 4) = make_float4(0.f, 0.f, 0.f, 0.f);
    }
  }

  int cur = 0;
  for (int c = 0; c < H_DIM / 64; ++c) {
    wait_async0();
    __syncthreads();                 // chunk c resident in xs[cur]
    if (c + 1 < H_DIM / 64) {        // overlap next chunk's gather with WMMA
      float* buf = xs[cur ^ 1];
      int h0n = (c + 1) * 64;
      for (int idx = threadIdx.x; idx < 34 * 16; idx += 256) {
        int r = idx >> 4, c4 = idx & 15;
        if (s0 + r >= 2)
          async_load_b128(lds_off(buf + r * 68 + c4 * 4),
                          x + (size_t)(t0 - 2 + r) * H_DIM + h0n + c4 * 4);
        else
          *(float4*)(buf + r * 68 + c4 * 4) = make_float4(0.f, 0.f, 0.f, 0.f);
      }
    }
    const float* buf = xs[cur];
    int h0 = c * 64;
#pragma unroll 2
    for (int ks = 0; ks < 16; ++ks) {
      int kb = ks * 4;
      int kg = h0 + kb;
      v2f a0[2], a1[2], a2[2];
#pragma unroll
      for (int mt = 0; mt < 2; ++mt) {
        int r = mrow + mt * 16;
        a0[mt] = load_afrag(buf, 68, r + 0, kb, lane);  // x[t-2]
        a1[mt] = load_afrag(buf, 68, r + 1, kb, lane);  // x[t-1]
        a2[mt] = load_afrag(buf, 68, r + 2, kb, lane);  // x[t]
      }
#pragma unroll
      for (int nt = 0; nt < 4; ++nt) {
        int n0 = nb + nt * 16;
        v2f b0 = load_bfrag(wct + 0 * HI, I_DIM, kg, n0, lane);
        v2f b1 = load_bfrag(wct + 1 * HI, I_DIM, kg, n0, lane);
        v2f b2 = load_bfrag(wct + 2 * HI, I_DIM, kg, n0, lane);
        v2f bu = load_bfrag(sup, I_DIM, kg, n0, lane);
#pragma unroll
        for (int mt = 0; mt < 2; ++mt) {   // B-fragments reused across M-subtiles
          gacc[nt][mt] = wmma_f32(a0[mt], b0, gacc[nt][mt]);
          gacc[nt][mt] = wmma_f32(a1[mt], b1, gacc[nt][mt]);
          gacc[nt][mt] = wmma_f32(a2[mt], b2, gacc[nt][mt]);
          uacc[nt][mt] = wmma_f32(a2[mt], bu, uacc[nt][mt]);
        }
      }
    }
    cur ^= 1;
  }

  // act = silu(gate) * up  -> LDS
  {
    int col = lane & 15, half = lane >> 4;
#pragma unroll
    for (int nt = 0; nt < 4; ++nt) {
      int n0 = nb + nt * 16;
#pragma unroll
      for (int mt = 0; mt < 2; ++mt)
#pragma unroll
        for (int j = 0; j < 8; ++j) {
          int m = j + half * 8 + mt * 16;
          act[m * 516 + n0 + col] = silu(gacc[nt][mt][j]) * uacc[nt][mt][j];
        }
    }
  }
  __syncthreads();

  // down GEMM: y(32x1024) = act(32x512) @ sdn(512x1024)
  v8f dacc[8][2];
#pragma unroll
  for (int nt = 0; nt < 8; ++nt) { dacc[nt][0] = vzero8(); dacc[nt][1] = vzero8(); }
  int hb = wv * 128;
#pragma unroll 2
  for (int k = 0; k < I_DIM; k += 4) {
    v2f a0 = load_afrag(act, 516, mrow, k, lane);
    v2f a1 = load_afrag(act, 516, mrow + 16, k, lane);
#pragma unroll
    for (int nt = 0; nt < 8; ++nt) {
      v2f b = load_bfrag(sdn, H_DIM, k, hb + nt * 16, lane);
      dacc[nt][0] = wmma_f32(a0, b, dacc[nt][0]);
      dacc[nt][1] = wmma_f32(a1, b, dacc[nt][1]);
    }
  }

  // store shared + identity-expert contribution
  {
    int col = lane & 15, half = lane >> 4;
#pragma unroll
    for (int nt = 0; nt < 8; ++nt) {
      int n0 = hb + nt * 16;
#pragma unroll
      for (int mt = 0; mt < 2; ++mt)
#pragma unroll
        for (int j = 0; j < 8; ++j) {
          int m = j + half * 8 + mt * 16;
          int t = t0 + m;
          size_t o = (size_t)t * H_DIM + n0 + col;
          out[o] = dacc[nt][mt][j] + id_w[t] * x[o];
        }
    }
  }
}

// ------------------------------------------------------------------
// 4) routed experts, M=32 gathered tiles, fused SwiGLU, atomic scatter
// ------------------------------------------------------------------
__global__ __launch_bounds__(256)
void expert_kernel(const float* __restrict__ x, const float* __restrict__ gw,
                   const float* __restrict__ uw, const float* __restrict__ dw,
                   const int* __restrict__ counts, const int* __restrict__ etok,
                   const float* __restrict__ ewt, float* __restrict__ out) {
  int e = blockIdx.y;
  int cnt = counts[e];
  int r0 = blockIdx.x * 32;
  if (r0 >= cnt) return;          // block-uniform: EXEC stays all-ones for WMMA

  __shared__ float xs[2][32 * 68];
  __shared__ float act[32 * 516];
  __shared__ int   tok_s[32];
  __shared__ float w_s[32];

  int lane = threadIdx.x & 31;
  int wv   = threadIdx.x >> 5;
  int mrow = lane & 15;

  if (threadIdx.x < 32) {
    int m = threadIdx.x;
    int valid = (r0 + m) < cnt;
    tok_s[m] = etok[(size_t)e * T_TOK + (valid ? r0 + m : r0)];
    w_s[m]   = valid ? ewt[(size_t)e * T_TOK + r0 + m] : 0.f;
  }
  __syncthreads();

  const float* gwe = gw + (size_t)e * HI;
  const float* uwe = uw + (size_t)e * HI;
  const float* dwe = dw + (size_t)e * HI;

  v8f gacc[4][2], uacc[4][2];
#pragma unroll
  for (int nt = 0; nt < 4; ++nt)
#pragma unroll
    for (int mt = 0; mt < 2; ++mt) { gacc[nt][mt] = vzero8(); uacc[nt][mt] = vzero8(); }
  int nb = wv * 64;

  // prologue: stage gathered chunk 0
  {
    float* buf = xs[0];
    for (int idx = threadIdx.x; idx < 32 * 16; idx += 256) {
      int r = idx >> 4, c4 = idx & 15;
      async_load_b128(lds_off(buf + r * 68 + c4 * 4),
                      x + (size_t)tok_s[r] * H_DIM + c4 * 4);
    }
  }

  int cur = 0;
  for (int c = 0; c < H_DIM / 64; ++c) {
    wait_async0();
    __syncthreads();
    if (c + 1 < H_DIM / 64) {
      float* buf = xs[cur ^ 1];
      int h0n = (c + 1) * 64;
      for (int idx = threadIdx.x; idx < 32 * 16; idx += 256) {
        int r = idx >> 4, c4 = idx & 15;
        async_load_b128(lds_off(buf + r * 68 + c4 * 4),
                        x + (size_t)tok_s[r] * H_DIM + h0n + c4 * 4);
      }
    }
    const float* buf = xs[cur];
    int h0 = c * 64;
#pragma unroll 2
    for (int ks = 0; ks < 16; ++ks) {
      int kb = ks * 4;
      int kg = h0 + kb;
      v2f a[2];
      a[0] = load_afrag(buf, 68, mrow, kb, lane);
      a[1] = load_afrag(buf, 68, mrow + 16, kb, lane);
#pragma unroll
      for (int nt = 0; nt < 4; ++nt) {
        int n0 = nb + nt * 16;
        v2f bg = load_bfrag(gwe, I_DIM, kg, n0, lane);
        v2f bu = load_bfrag(uwe, I_DIM, kg, n0, lane);
#pragma unroll
        for (int mt = 0; mt < 2; ++mt) {
          gacc[nt][mt] = wmma_f32(a[mt], bg, gacc[nt][mt]);
          uacc[nt][mt] = wmma_f32(a[mt], bu, uacc[nt][mt]);
        }
      }
    }
    cur ^= 1;
  }

  {
    int col = lane & 15, half = lane >> 4;
#pragma unroll
    for (int nt = 0; nt < 4; ++nt) {
      int n0 = nb + nt * 16;
#pragma unroll
      for (int mt = 0; mt < 2; ++mt)
#pragma unroll
        for (int j = 0; j < 8; ++j) {
          int m = j + half * 8 + mt * 16;
          act[m * 516 + n0 + col] = silu(gacc[nt][mt][j]) * uacc[nt][mt][j];
        }
    }
  }
  __syncthreads();

  v8f dacc[8][2];
#pragma unroll
  for (int nt = 0; nt < 8; ++nt) { dacc[nt][0] = vzero8(); dacc[nt][1] = vzero8(); }
  int hb = wv * 128;
#pragma unroll 2
  for (int k = 0; k < I_DIM; k += 4) {
    v2f a0 = load_afrag(act, 516, mrow, k, lane);
    v2f a1 = load_afrag(act, 516, mrow + 16, k, lane);
#pragma unroll
    for (int nt = 0; nt < 8; ++nt) {
      v2f b = load_bfrag(dwe, H_DIM, k, hb + nt * 16, lane);
      dacc[nt][0] = wmma_f32(a0, b, dacc[nt][0]);
      dacc[nt][1] = wmma_f32(a1, b, dacc[nt][1]);
    }
  }

  {
    int col = lane & 15, half = lane >> 4;
#pragma unroll
    for (int nt = 0; nt < 8; ++nt) {
      int n0 = hb + nt * 16;
#pragma unroll
      for (int mt = 0; mt < 2; ++mt)
#pragma unroll
        for (int j = 0; j < 8; ++j) {
          int m = j + half * 8 + mt * 16;
          float wgt = w_s[m];
          if (wgt != 0.f)
            atomicAdd(&out[(size_t)tok_s[m] * H_DIM + n0 + col],
                      wgt * dacc[nt][mt][j]);
        }
    }
  }
}

// ------------------------------------------------------------------
extern "C" void kernel_launch(void* const* d_in, const int* in_sizes, int n_in,
                              void* d_out, int out_size, void* d_ws, size_t ws_size,
                              hipStream_t stream) {
  const float* x   = (const float*)d_in[0];
  const float* rw  = (const float*)d_in[1];
  const float* rb  = (const float*)d_in[2];
  const float* gw  = (const float*)d_in[3];
  const float* uw  = (const float*)d_in[4];
  const float* dwn = (const float*)d_in[5];
  const float* cw  = (const float*)d_in[6];
  const float* sup = (const float*)d_in[7];
  const float* sdn = (const float*)d_in[8];
  float* out = (float*)d_out;

  char* ws = (char*)d_ws;
  int*   counts = (int*)ws;                                  // 16 ints (pad 256B)
  float* id_w   = (float*)(ws + 256);                        // T floats
  int*   etok   = (int*)(ws + 256 + (size_t)T_TOK * 4);      // 15*T ints
  float* ewt    = (float*)(ws + 256 + (size_t)T_TOK * 4 + (size_t)E_R * T_TOK * 4);
  float* wct    = (float*)(ws + 256 + (size_t)T_TOK * 4 + 2 * (size_t)E_R * T_TOK * 4);
  // wct: 3*H*I floats; total ws use ~8.3 MB

  init_counts_kernel<<<1, 64, 0, stream>>>(counts);
  conv_transpose_kernel<<<(HI + 255) / 256, 256, 0, stream>>>(cw, wct);
  router_kernel<<<T_TOK / 8, 256, 0, stream>>>(x, rw, rb, counts, id_w, etok, ewt);
  shared_kernel<<<T_TOK / 32, 256, 0, stream>>>(x, wct, sup, sdn, id_w, out);
  dim3 egrid(T_TOK / 32, E_R);
  expert_kernel<<<egrid, 256, 0, stream>>>(x, gw, uw, dwn, counts, etok, ewt, out);
}